// _multiHead_GAT_88811333747472
// MI455X (gfx1250) — compile-verified
//
#include <hip/hip_runtime.h>
#include <math.h>

// GAT layer for MI455X (gfx1250).
// Pipeline:
//   prep:   pack Wq -> Wbig[72x80], Wl -> block-diag WsBig/WrBig[72x80], biases (pad cols 72..79 = 0)
//   gemm:   q  = nodes x Wbig + bq          (WMMA f32 16x16x4)
//           qs = q x WsBig                  (WMMA)
//           qr = q x WrBig + bl             (WMMA)
//   init:   maxkey = ord(-inf), denom = 0, agg = 0        (vectorized b128)
//   passA:  per (edge,c4): logit4 = qs4[s]+qr4[r]; atomicMax(maxkey) x4      (b128 gathers)
//   passB:  w = exp(logit-max); atomicAdd(denom,w); atomicAdd(agg, q[s]*w)   (b128 gathers)
//   final:  out = nodes + leaky_relu(agg/denom)   (denom==0 -> 0, empty segment)

#define DM   72
#define D4   18          // DM / 4 float4 chunks per row
#define HH   6
#define FF   12
#define PADC 80

typedef __attribute__((ext_vector_type(2))) float v2f;
typedef __attribute__((ext_vector_type(8))) float v8f;

// ---- monotone float<->int key for integer atomicMax on floats ----
__device__ __forceinline__ int f2ord(float f) {
    int i = __float_as_int(f);
    return i >= 0 ? i : (i ^ 0x7FFFFFFF);
}
__device__ __forceinline__ float ord2f(int i) {
    return __int_as_float(i >= 0 ? i : (i ^ 0x7FFFFFFF));
}

// ---------------- prep: pack weights into padded [72 x 80] row-major ----------------
__global__ void gat_prep_kernel(const float* __restrict__ Wq,  // [H,D,F]
                                const float* __restrict__ bq,  // [H,F]
                                const float* __restrict__ Wl,  // [H,2F,F]
                                const float* __restrict__ bl,  // [H,F]
                                float* __restrict__ Wbig,      // [72,80]
                                float* __restrict__ WsBig,     // [72,80]
                                float* __restrict__ WrBig,     // [72,80]
                                float* __restrict__ bqPad,     // [80]
                                float* __restrict__ blPad,     // [80]
                                float* __restrict__ zeroPad) { // [80]
    for (int idx = threadIdx.x; idx < DM * PADC; idx += blockDim.x) {
        int k = idx / PADC;     // input dim row 0..71
        int c = idx % PADC;     // output col 0..79
        float wq = 0.f, ws = 0.f, wr = 0.f;
        if (c < DM) {
            int h = c / FF, f = c % FF;
            wq = Wq[(h * DM + k) * FF + f];          // Wq[h][k][f]
            int hk = k / FF, fp = k % FF;
            if (hk == h) {                            // block-diagonal
                ws = Wl[(h * 2 * FF + fp) * FF + f];      // Wl[h][fp][f]
                wr = Wl[(h * 2 * FF + FF + fp) * FF + f]; // Wl[h][F+fp][f]
            }
        }
        Wbig[idx] = wq;
        WsBig[idx] = ws;
        WrBig[idx] = wr;
    }
    for (int c = threadIdx.x; c < PADC; c += blockDim.x) {
        bqPad[c]  = (c < DM) ? bq[c] : 0.f;
        blPad[c]  = (c < DM) ? bl[c] : 0.f;
        zeroPad[c] = 0.f;
    }
}

// ---------------- WMMA GEMM: D[M x 72] = A[M x 72] * B[72 x 80] + bias ----------------
// One wave computes one 16x16 output tile via 18 chained V_WMMA_F32_16X16X4_F32.
// Fragment layouts per CDNA5 ISA 7.12.2:
//   A 16x4:  lanes 0-15 hold K=0,1 (v0,v1); lanes 16-31 hold K=2,3
//   B 4x16:  N = lane&15; lanes 0-15 hold K=0,1; lanes 16-31 hold K=2,3
//   D 16x16: N = lane&15; VGPR j holds M = j (lanes 0-15) / M = 8+j (lanes 16-31)
__global__ void gat_gemm_wmma(const float* __restrict__ A,
                              const float* __restrict__ B,
                              const float* __restrict__ bias,
                              float* __restrict__ D,
                              int M) {
    const int lane    = threadIdx.x;                       // 0..31
    const int rowTile = blockIdx.y * blockDim.y + threadIdx.y;
    const int colTile = blockIdx.x;                        // 0..4
    const int row0 = rowTile * 16;
    const int col0 = colTile * 16;
    const int lo   = lane & 15;
    const int hi   = lane >> 4;                            // 0 or 1
    const int kb   = hi * 2;                               // K sub-offset 0 or 2
    const int col  = col0 + lo;

    // clamp A row so loads stay in-bounds without diverging EXEC for WMMA
    int arow = row0 + lo;
    if (arow >= M) arow = M - 1;
    const float* Arow = A + (size_t)arow * DM;

    v8f acc = {};
#pragma unroll
    for (int k0 = 0; k0 < DM; k0 += 4) {
        v2f a, b;
        a.x = Arow[k0 + kb];
        a.y = Arow[k0 + kb + 1];
        b.x = B[(size_t)(k0 + kb) * PADC + col];       // cols 72..79 are zero-padded
        b.y = B[(size_t)(k0 + kb + 1) * PADC + col];
        acc = __builtin_amdgcn_wmma_f32_16x16x4_f32(
            /*neg_a=*/false, a, /*neg_b=*/false, b,
            /*c_mod=*/(short)0, acc, /*reuse_a=*/false, /*reuse_b=*/false);
    }

    if (col < DM) {
        const float bv = bias[col];
        const int rbase = row0 + (hi << 3);            // +8 for upper-half lanes
#pragma unroll
        for (int j = 0; j < 8; ++j) {
            int r = rbase + j;
            if (r < M) D[(size_t)r * DM + col] = acc[j] + bv;
        }
    }
}

// ---------------- per-call reinit of accumulators (b128 stores) ----------------
__global__ void gat_init_kernel(int4* __restrict__ maxk4,
                                float4* __restrict__ denom4,
                                float4* __restrict__ agg4,
                                int total4) {              // N * D4
    int idx = blockIdx.x * blockDim.x + threadIdx.x;
    if (idx < total4) {
        const int NEG = (int)0x807FFFFF;                   // f2ord(-inf)
        maxk4[idx]  = make_int4(NEG, NEG, NEG, NEG);
        denom4[idx] = make_float4(0.f, 0.f, 0.f, 0.f);
        agg4[idx]   = make_float4(0.f, 0.f, 0.f, 0.f);
    }
}

// ---------------- edge pass A: segment max of logits (float4 chunks) ----------------
__global__ void gat_edge_max(const float4* __restrict__ qs4,
                             const float4* __restrict__ qr4,
                             const int* __restrict__ snd,
                             const int* __restrict__ rcv,
                             int* __restrict__ maxk,
                             int total) {                  // E * D4
    int idx = blockIdx.x * blockDim.x + threadIdx.x;
    if (idx >= total) return;
    int e  = idx / D4;
    int c4 = idx % D4;
    int s = snd[e], r = rcv[e];
    float4 a = qs4[(size_t)s * D4 + c4];
    float4 b = qr4[(size_t)r * D4 + c4];
    int* mk = maxk + ((size_t)r * D4 + c4) * 4;
    atomicMax(mk + 0, f2ord(a.x + b.x));
    atomicMax(mk + 1, f2ord(a.y + b.y));
    atomicMax(mk + 2, f2ord(a.z + b.z));
    atomicMax(mk + 3, f2ord(a.w + b.w));
}

// ---------------- edge pass B: exp-weights + weighted message aggregation ----------------
__global__ void gat_edge_sum(const float4* __restrict__ q4,
                             const float4* __restrict__ qs4,
                             const float4* __restrict__ qr4,
                             const int* __restrict__ snd,
                             const int* __restrict__ rcv,
                             const int4* __restrict__ maxk4,
                             float* __restrict__ denom,
                             float* __restrict__ agg,
                             int total) {                  // E * D4
    int idx = blockIdx.x * blockDim.x + threadIdx.x;
    if (idx >= total) return;
    int e  = idx / D4;
    int c4 = idx % D4;
    int s = snd[e], r = rcv[e];
    size_t soff = (size_t)s * D4 + c4;
    size_t roff = (size_t)r * D4 + c4;
    float4 a  = qs4[soff];
    float4 b  = qr4[roff];
    float4 sv = q4[soff];
    int4   mk = maxk4[roff];
    float w0 = __expf(a.x + b.x - ord2f(mk.x));
    float w1 = __expf(a.y + b.y - ord2f(mk.y));
    float w2 = __expf(a.z + b.z - ord2f(mk.z));
    float w3 = __expf(a.w + b.w - ord2f(mk.w));
    float* dn = denom + roff * 4;
    float* ag = agg   + roff * 4;
    atomicAdd(dn + 0, w0);
    atomicAdd(dn + 1, w1);
    atomicAdd(dn + 2, w2);
    atomicAdd(dn + 3, w3);
    atomicAdd(ag + 0, sv.x * w0);
    atomicAdd(ag + 1, sv.y * w1);
    atomicAdd(ag + 2, sv.z * w2);
    atomicAdd(ag + 3, sv.w * w3);
}

// ---------------- node finalize: residual + leaky_relu (b128) ----------------
__device__ __forceinline__ float lrelu_div(float a, float d) {
    float v = (d > 0.f) ? (a / d) : 0.f;                   // empty segment -> 0
    return (v > 0.f) ? v : 0.01f * v;                      // leaky_relu, slope 0.01
}

__global__ void gat_final(const float4* __restrict__ nodes4,
                          const float4* __restrict__ denom4,
                          const float4* __restrict__ agg4,
                          float4* __restrict__ out4,
                          int total4) {                    // N * D4
    int idx = blockIdx.x * blockDim.x + threadIdx.x;
    if (idx >= total4) return;
    float4 n = nodes4[idx];
    float4 d = denom4[idx];
    float4 a = agg4[idx];
    float4 o;
    o.x = n.x + lrelu_div(a.x, d.x);
    o.y = n.y + lrelu_div(a.y, d.y);
    o.z = n.z + lrelu_div(a.z, d.z);
    o.w = n.w + lrelu_div(a.w, d.w);
    out4[idx] = o;
}

extern "C" void kernel_launch(void* const* d_in, const int* in_sizes, int n_in,
                              void* d_out, int out_size, void* d_ws, size_t ws_size,
                              hipStream_t stream) {
    const float* nodes = (const float*)d_in[0];   // [N,72]
    const int*   snd   = (const int*)d_in[1];     // [E]
    const int*   rcv   = (const int*)d_in[2];     // [E]
    const float* Wq    = (const float*)d_in[3];   // [H,D,F]
    const float* bq    = (const float*)d_in[4];   // [H,F]
    const float* Wl    = (const float*)d_in[5];   // [H,2F,F]
    const float* bl    = (const float*)d_in[6];   // [H,F]
    float* out = (float*)d_out;

    const int N = in_sizes[0] / DM;
    const int E = in_sizes[1];

    // ---- workspace carve-out (256B aligned) ----
    char* base = (char*)d_ws;
    size_t off = 0;
    auto carve = [&](size_t bytes) -> char* {
        char* p = base + off;
        off = (off + bytes + 255) & ~(size_t)255;
        return p;
    };
    float* Wbig  = (float*)carve(DM * PADC * sizeof(float));
    float* WsBig = (float*)carve(DM * PADC * sizeof(float));
    float* WrBig = (float*)carve(DM * PADC * sizeof(float));
    float* bqPad = (float*)carve(PADC * sizeof(float));
    float* blPad = (float*)carve(PADC * sizeof(float));
    float* zPad  = (float*)carve(PADC * sizeof(float));
    float* q     = (float*)carve((size_t)N * DM * sizeof(float));
    float* qs    = (float*)carve((size_t)N * DM * sizeof(float));
    float* qr    = (float*)carve((size_t)N * DM * sizeof(float));
    int*   maxk  = (int*)  carve((size_t)N * DM * sizeof(int));
    float* denom = (float*)carve((size_t)N * DM * sizeof(float));
    float* agg   = (float*)carve((size_t)N * DM * sizeof(float));
    (void)ws_size;

    // 1) pack weights
    gat_prep_kernel<<<1, 256, 0, stream>>>(Wq, bq, Wl, bl, Wbig, WsBig, WrBig,
                                           bqPad, blPad, zPad);

    // 2) node-level GEMMs via WMMA f32 16x16x4
    const int rowTiles = (N + 15) / 16;                 // 6250 for N=100000
    dim3 gblk(32, 2);
    dim3 ggrd((DM + 15) / 16, (rowTiles + 1) / 2);      // (5, 3125)
    gat_gemm_wmma<<<ggrd, gblk, 0, stream>>>(nodes, Wbig,  bqPad, q,  N);
    gat_gemm_wmma<<<ggrd, gblk, 0, stream>>>(q,     WsBig, zPad,  qs, N);
    gat_gemm_wmma<<<ggrd, gblk, 0, stream>>>(q,     WrBig, blPad, qr, N);

    // 3) reset accumulators (required every call: graph replay reuses ws)
    const int nTot4 = N * D4;
    gat_init_kernel<<<(nTot4 + 255) / 256, 256, 0, stream>>>(
        (int4*)maxk, (float4*)denom, (float4*)agg, nTot4);

    // 4) segment softmax + aggregation over edges (float4 chunks)
    const int eTot4 = E * D4;                           // 28.8M for E=1.6M
    const int eBlocks = (eTot4 + 255) / 256;
    gat_edge_max<<<eBlocks, 256, 0, stream>>>(
        (const float4*)qs, (const float4*)qr, snd, rcv, maxk, eTot4);
    gat_edge_sum<<<eBlocks, 256, 0, stream>>>(
        (const float4*)q, (const float4*)qs, (const float4*)qr, snd, rcv,
        (const int4*)maxk, denom, agg, eTot4);

    // 5) finalize
    gat_final<<<(nTot4 + 255) / 256, 256, 0, stream>>>(
        (const float4*)nodes, (const float4*)denom, (const float4*)agg,
        (float4*)out, nTot4);
}